// CandidatePenaltyCrossEntropyCriterion_65060164599830
// MI455X (gfx1250) — compile-verified
//
#include <hip/hip_runtime.h>
#include <stdint.h>
#include <math.h>

// Problem constants (match reference)
#define BSZ 2
#define SEQ 1024
#define VOCAB 50257
#define IGNORE_INDEX (-100)
#define CLAMP_MIN 1e-5f

#define NROWS (BSZ * SEQ)          // 2048 rows
#define VPAD 50260                 // VOCAB rounded up to multiple of 4
#define NCHUNK (VPAD / 4)          // 12565 float4 chunks per row (last one partial)
#define WG 1024                    // 32 wave32 waves per workgroup
#define NWAVES (WG / 32)
#define KFULL 12                   // chunks 0..11 are uniform across all 1024 threads
#define LAST_B128_TID 275          // k=12: tid<=275 full b128 chunks (idx 12288..12563)
#define TAIL_TID 276               // tid==276 owns partial chunk 12564 (elem 50256 + pad)
#define RED_SLOTS 64
#define DYN_LDS_BYTES (((size_t)VPAD + RED_SLOTS) * sizeof(float))  // ~201.3 KB < 320 KB/WGP

// ---------------------------------------------------------------------------
// CDNA5 async global->LDS DMA path (ASYNCcnt-tracked, gfx1250)
// ---------------------------------------------------------------------------
__device__ __forceinline__ void async_g2lds_b128(uint32_t lds_off, const void* gptr) {
  asm volatile("global_load_async_to_lds_b128 %0, %1, off"
               :: "v"(lds_off), "v"((uint64_t)gptr) : "memory");
}
__device__ __forceinline__ void async_g2lds_b32(uint32_t lds_off, const void* gptr) {
  asm volatile("global_load_async_to_lds_b32 %0, %1, off"
               :: "v"(lds_off), "v"((uint64_t)gptr) : "memory");
}
template <int N>
__device__ __forceinline__ void wait_async_le() {
  asm volatile("s_wait_asynccnt %0" :: "i"(N) : "memory");
}

__device__ __forceinline__ void online_update(float& m, float& sum, float4 x) {
  float cm = fmaxf(fmaxf(x.x, x.y), fmaxf(x.z, x.w));
  if (cm > m) { sum *= __expf(m - cm); m = cm; }
  sum += __expf(x.x - m) + __expf(x.y - m) + __expf(x.z - m) + __expf(x.w - m);
}
__device__ __forceinline__ void merge_ms(float& m, float& s, float om, float os) {
  float nm = fmaxf(m, om);
  s = s * __expf(m - nm) + os * __expf(om - nm);
  m = nm;
}

// Pipelined pass 1 over chunks 0..11: ASYNCcnt decrements in issue order, and
// every wave issues >= 12 async ops, so asynccnt <= (11-K) guarantees this
// wave's K-th issued chunk has landed in LDS. Compute overlaps the HBM stream.
template <int K>
__device__ __forceinline__ void pass1_steps(const float4* sv, int tid, float& m, float& sum) {
  wait_async_le<KFULL - 1 - K>();
  online_update(m, sum, sv[tid + K * WG]);
  if constexpr (K + 1 < KFULL) pass1_steps<K + 1>(sv, tid, m, sum);
}

// ---------------------------------------------------------------------------
// Kernel 1: first_occ32[b][v] = INT_MAX (ws is dirty each call)
// ---------------------------------------------------------------------------
__global__ void fo_init_kernel(int* __restrict__ fo32) {
  int i = blockIdx.x * blockDim.x + threadIdx.x;
  if (i < BSZ * VPAD) fo32[i] = 0x7FFFFFFF;
}

// ---------------------------------------------------------------------------
// Kernel 2: first occurrence position of each valid token per batch
// (atomicMin is order-independent -> deterministic)
// ---------------------------------------------------------------------------
__global__ void fo_scatter_kernel(const int* __restrict__ target, int* __restrict__ fo32) {
  int i = blockIdx.x * blockDim.x + threadIdx.x;
  if (i < NROWS) {
    int t = target[i];
    if (t != IGNORE_INDEX) atomicMin(&fo32[(i / SEQ) * VPAD + t], i % SEQ);
  }
}

// ---------------------------------------------------------------------------
// Kernel 3: pack to u16 (positions <= 1023; sentinel 0xFFFF) -> halves the
// L2-resident side stream read by every row
// ---------------------------------------------------------------------------
__global__ void fo_pack_kernel(const int* __restrict__ fo32, unsigned short* __restrict__ fo16) {
  int i = blockIdx.x * blockDim.x + threadIdx.x;
  if (i < BSZ * VPAD) {
    int v = fo32[i];
    fo16[i] = (unsigned short)(v > 0xFFFF ? 0xFFFF : v);
  }
}

// ---------------------------------------------------------------------------
// Kernel 4: one workgroup per (b,s) row.
//   - issue async DMA of the 50257-float row into LDS (single HBM pass)
//   - prefetch this thread's 13 u16x4 first-occ words into registers
//   - pass 1: per-chunk s_wait_asynccnt pipelining, online max/sum-exp
//     (no barrier: each thread consumes exactly the chunks it issued)
//   - pass 2: MLE + unlikelihood penalty from LDS + registers
// ---------------------------------------------------------------------------
__global__ __launch_bounds__(WG) void row_loss_kernel(
    const float* __restrict__ logits, const int* __restrict__ target,
    const unsigned short* __restrict__ fo16, float* __restrict__ partials) {
  extern __shared__ __align__(16) float smem[];  // VPAD row values + RED_SLOTS scratch

  const int tid = threadIdx.x;
  const int row = blockIdx.x;
  const int b = row >> 10;          // row / SEQ
  const int s = row & (SEQ - 1);    // row % SEQ
  const int tgt = target[row];
  const float* __restrict__ rowp = logits + (size_t)row * VOCAB;
  const uint32_t lds0 = __builtin_amdgcn_groupstaticsize();  // dynamic LDS base offset

  // ---- issue the whole row as async DMA (every wave issues 12..14 ops) ----
  #pragma unroll
  for (int k = 0; k < KFULL; ++k) {
    int idx = tid + k * WG;
    async_g2lds_b128(lds0 + (uint32_t)idx * 16u, rowp + (size_t)idx * 4);
  }
  if (tid <= LAST_B128_TID) {
    int idx = tid + KFULL * WG;
    async_g2lds_b128(lds0 + (uint32_t)idx * 16u, rowp + (size_t)idx * 4);
  }
  if (tid == TAIL_TID) {
    async_g2lds_b32(lds0 + (uint32_t)(VOCAB - 1) * 4u, rowp + (VOCAB - 1));
    // -inf padding written by the SAME thread that later reads chunk 12564
    smem[VOCAB + 0] = -INFINITY;
    smem[VOCAB + 1] = -INFINITY;
    smem[VOCAB + 2] = -INFINITY;
  }

  // ---- prefetch first-occ u16 words into registers (flies under the DMA) ----
  const uint64_t* fo8 = (const uint64_t*)(fo16 + (size_t)b * VPAD);  // 4 tokens / 8 B
  uint64_t fo_reg[KFULL + 1];
  #pragma unroll
  for (int k = 0; k < KFULL; ++k) fo_reg[k] = fo8[tid + k * WG];
  fo_reg[KFULL] = (tid <= TAIL_TID) ? fo8[tid + KFULL * WG] : 0xFFFFFFFFFFFFFFFFull;

  const int lane = tid & 31;
  const int wave = tid >> 5;
  float* redm = smem + VPAD;
  float* reds = redm + NWAVES;
  const float4* sv = (const float4*)smem;

  // ---- pass 1: pipelined online max / sum-exp ----
  float m = -INFINITY, sum = 0.0f;
  pass1_steps<0>(sv, tid, m, sum);
  wait_async_le<0>();               // last 1-2 chunks (k=12 + b32 tail)
  if (tid <= TAIL_TID) online_update(m, sum, sv[tid + KFULL * WG]);

  for (int off = 16; off > 0; off >>= 1)
    merge_ms(m, sum, __shfl_xor(m, off, 32), __shfl_xor(sum, off, 32));
  if (lane == 0) { redm[wave] = m; reds[wave] = sum; }
  __syncthreads();
  if (wave == 0) {
    float wm = redm[lane], ws = reds[lane];
    for (int off = 16; off > 0; off >>= 1)
      merge_ms(wm, ws, __shfl_xor(wm, off, 32), __shfl_xor(ws, off, 32));
    if (lane == 0) { redm[0] = wm; reds[0] = ws; }
  }
  __syncthreads();
  const float row_lse = redm[0] + __logf(reds[0]);
  __syncthreads();  // lse consumed into registers; red slots reusable

  // ---- pass 2: MLE + unlikelihood penalty (LDS + registers only) ----
  float acc = 0.0f;
  #pragma unroll
  for (int k = 0; k <= KFULL; ++k) {
    if (k == KFULL && tid > TAIL_TID) break;
    int idx = tid + k * WG;
    float4 x = sv[idx];
    uint64_t f = fo_reg[k];
    int vb = idx * 4;
    #pragma unroll
    for (int j = 0; j < 4; ++j) {
      float xv = (j == 0) ? x.x : (j == 1) ? x.y : (j == 2) ? x.z : x.w;
      int fv = (int)((f >> (16 * j)) & 0xFFFFu);   // padded slots = 0xFFFF -> never seen
      int v = vb + j;
      float lp = xv - row_lse;
      if (v == tgt) {
        acc -= lp;                                  // IGNORE_INDEX (<0) never matches
      } else if (fv < s) {
        float omp = fmaxf(1.0f - __expf(lp), CLAMP_MIN);
        acc -= __logf(omp);
      }
    }
  }
  for (int off = 16; off > 0; off >>= 1) acc += __shfl_xor(acc, off, 32);
  if (lane == 0) redm[wave] = acc;
  __syncthreads();
  if (wave == 0) {
    float a = redm[lane];
    for (int off = 16; off > 0; off >>= 1) a += __shfl_xor(a, off, 32);
    if (lane == 0) partials[row] = a;
  }
}

// ---------------------------------------------------------------------------
// Kernel 5: sum 2048 per-row partials, count valid targets, write loss/weight
// ---------------------------------------------------------------------------
__global__ __launch_bounds__(WG) void finalize_kernel(
    const float* __restrict__ partials, const int* __restrict__ target,
    float* __restrict__ out) {
  __shared__ float sf[NWAVES];
  __shared__ float sw[NWAVES];
  float a = 0.0f, w = 0.0f;
  for (int i = threadIdx.x; i < NROWS; i += WG) {
    a += partials[i];
    w += (target[i] != IGNORE_INDEX) ? 1.0f : 0.0f;
  }
  int lane = threadIdx.x & 31, wave = threadIdx.x >> 5;
  for (int off = 16; off > 0; off >>= 1) {
    a += __shfl_xor(a, off, 32);
    w += __shfl_xor(w, off, 32);
  }
  if (lane == 0) { sf[wave] = a; sw[wave] = w; }
  __syncthreads();
  if (wave == 0) {
    a = sf[lane]; w = sw[lane];
    for (int off = 16; off > 0; off >>= 1) {
      a += __shfl_xor(a, off, 32);
      w += __shfl_xor(w, off, 32);
    }
    if (lane == 0) out[0] = a / w;
  }
}

// ---------------------------------------------------------------------------
extern "C" void kernel_launch(void* const* d_in, const int* in_sizes, int n_in,
                              void* d_out, int out_size, void* d_ws, size_t ws_size,
                              hipStream_t stream) {
  const float* logits = (const float*)d_in[0];   // [2,1024,50257] fp32
  const int* target = (const int*)d_in[1];       // [2,1024] int32
  float* out = (float*)d_out;                    // scalar

  // workspace layout (all offsets naturally aligned)
  int* fo32 = (int*)d_ws;                                                  // 402,080 B
  unsigned short* fo16 = (unsigned short*)((char*)d_ws + (size_t)BSZ * VPAD * 4);  // 201,040 B
  float* partials = (float*)((char*)d_ws + (size_t)BSZ * VPAD * 6);        // 8,192 B

  fo_init_kernel<<<(BSZ * VPAD + 255) / 256, 256, 0, stream>>>(fo32);
  fo_scatter_kernel<<<(NROWS + 255) / 256, 256, 0, stream>>>(target, fo32);
  fo_pack_kernel<<<(BSZ * VPAD + 255) / 256, 256, 0, stream>>>(fo32, fo16);
  row_loss_kernel<<<NROWS, WG, DYN_LDS_BYTES, stream>>>(logits, target, fo16, partials);
  finalize_kernel<<<1, WG, 0, stream>>>(partials, target, out);
}